// GaranAttentionV2_50740743635260
// MI455X (gfx1250) — compile-verified
//
#include <hip/hip_runtime.h>
#include <hip/hip_bf16.h>

typedef float v8f __attribute__((ext_vector_type(8)));
typedef float v2f __attribute__((ext_vector_type(2)));

// ---------------------------------------------------------------------------
// Problem constants (from reference): b=32, l=64, d_q=1024, c=d_o=512,
// h=w=32 (hw=1024), n_head=8, dk=64, temp=sqrt(64)=8.
// ---------------------------------------------------------------------------
#define NB   32
#define LQ   64
#define DQ   1024
#define CH   512
#define HW   1024
#define NH   8
#define DK   64
#define INVTEMP 0.125f

// ---------------------------------------------------------------------------
// K1: qh_sum[b,o] = (sum_l q[b,l,:]) @ Wqs^T + 64*bqs   (flap==1 identity)
// ---------------------------------------------------------------------------
__global__ __launch_bounds__(256)
void k_qsum_proj(const float* __restrict__ q, const float* __restrict__ Wqs,
                 const float* __restrict__ bqs, float* __restrict__ qh_sum) {
  __shared__ float qs[DQ];
  const int b = blockIdx.x, t = threadIdx.x;
  const float* qb = q + (size_t)b * LQ * DQ;
  for (int i = t; i < DQ; i += 256) {
    float s = 0.f;
    for (int l = 0; l < LQ; ++l) s += qb[(size_t)l * DQ + i];
    qs[i] = s;
  }
  __syncthreads();
  for (int o = t; o < CH; o += 256) {
    const float* wr = Wqs + (size_t)o * DQ;
    float acc = 0.f;
    for (int i = 0; i < DQ; ++i) acc += qs[i] * wr[i];
    qh_sum[b * CH + o] = acc + 64.0f * bqs[o];
  }
}

// ---------------------------------------------------------------------------
// K2: fold conv weights through qh_sum.
//   rows 0..7 : u_c[b,n,c] = sum_d qh_sum[b,n*64+d]*Wkc[n*64+d,c]
//   rows 8..15: u_d via Wkd.   kconst[b,r] = sum_d qh_sum*bias.
// ---------------------------------------------------------------------------
__global__ __launch_bounds__(128)
void k_fold(const float* __restrict__ qh_sum,
            const float* __restrict__ wkc, const float* __restrict__ bkc,
            const float* __restrict__ wkd, const float* __restrict__ bkd,
            float* __restrict__ U, float* __restrict__ kconst) {
  const int r = blockIdx.x;   // 0..15
  const int b = blockIdx.y;   // 0..31
  const int n = r & 7;
  const float* w    = (r >= 8) ? wkd : wkc;
  const float* bias = (r >= 8) ? bkd : bkc;
  __shared__ float qh[DK];
  __shared__ float red[128];
  const int t = threadIdx.x;
  if (t < DK) qh[t] = qh_sum[b * CH + n * DK + t];
  __syncthreads();
  for (int c = t; c < CH; c += 128) {
    float acc = 0.f;
    for (int d = 0; d < DK; ++d) acc += qh[d] * w[(size_t)(n * DK + d) * CH + c];
    U[((size_t)b * 16 + r) * CH + c] = acc;
  }
  red[t] = (t < DK) ? qh[t] * bias[n * DK + t] : 0.f;
  __syncthreads();
  for (int s = 64; s > 0; s >>= 1) { if (t < s) red[t] += red[t + s]; __syncthreads(); }
  if (t == 0) kconst[b * 16 + r] = red[0];
}

// ---------------------------------------------------------------------------
// K3 (WMMA): S[b,16,1024] = U[b,16,512] @ v[b,512,1024], then (+ const) * 1/temp.
// One wave per 16x16 tile; V_WMMA_F32_16X16X4_F32, K-loop of 128 steps.
// A (16x4 f32): lane<16 -> M=lane,K={0,1}; lane>=16 -> M=lane-16,K={2,3}.
// B (4x16 f32): lane<16 -> N=lane,K={0,1}; lane>=16 -> N=lane-16,K={2,3}.
// D: VGPR r, lanes 0-15 = row r (score_c head r); lanes 16-31 = row r+8
//    (score_d head r). EXEC is all-ones (32-thread block, no divergence).
// ---------------------------------------------------------------------------
__global__ __launch_bounds__(32)
void k_score_wmma(const float* __restrict__ v, const float* __restrict__ U,
                  const float* __restrict__ kconst, float* __restrict__ sc) {
  const int tile = blockIdx.x;            // 0..63
  const int b    = blockIdx.y;            // 0..31
  const int lane = threadIdx.x;           // 0..31
  const int s0   = tile * 16;
  const int row  = lane & 15;
  const int hi   = lane >> 4;             // 0 or 1

  const float* Ub = U + (size_t)b * 16 * CH;
  const float* vb = v + (size_t)b * CH * HW;
  const float* aptr = Ub + (size_t)row * CH + 2 * hi;
  const float* bptr = vb + (size_t)(2 * hi) * HW + s0 + row;

  v8f acc = {};
#pragma unroll 4
  for (int k = 0; k < CH; k += 4) {
    v2f a = *reinterpret_cast<const v2f*>(aptr + k);   // U[row][k+2hi .. +1]
    v2f bb;
    bb.x = bptr[(size_t)k * HW];                        // v[k+2hi  ][s0+row]
    bb.y = bptr[(size_t)k * HW + HW];                   // v[k+2hi+1][s0+row]
    acc = __builtin_amdgcn_wmma_f32_16x16x4_f32(
        /*neg_a=*/false, a, /*neg_b=*/false, bb,
        /*c_mod=*/(short)0, acc, /*reuse_a=*/false, /*reuse_b=*/false);
  }
#pragma unroll
  for (int r = 0; r < 8; ++r) {
    const int vr = r + hi * 8;                          // output row 0..15
    const float kc = kconst[b * 16 + vr];
    sc[((size_t)b * 16 + vr) * HW + s0 + row] = (acc[r] + kc) * INVTEMP;
  }
}

// ---------------------------------------------------------------------------
// K4: p[b,n,:] = softmax(sc[b,n,:]) ; sig[b,n,:] = sigmoid(sc[b,8+n,:])
// ---------------------------------------------------------------------------
__global__ __launch_bounds__(256)
void k_softmax_sig(const float* __restrict__ sc, float* __restrict__ p,
                   float* __restrict__ sig) {
  const int n = blockIdx.x, b = blockIdx.y, t = threadIdx.x;
  const float* rc = sc + ((size_t)b * 16 + n) * HW;
  const float* rd = sc + ((size_t)b * 16 + 8 + n) * HW;
  __shared__ float red[256];
  float m = -3.402823466e+38f;
  for (int i = t; i < HW; i += 256) m = fmaxf(m, rc[i]);
  red[t] = m; __syncthreads();
  for (int s = 128; s > 0; s >>= 1) { if (t < s) red[t] = fmaxf(red[t], red[t + s]); __syncthreads(); }
  const float mx = red[0]; __syncthreads();
  float sum = 0.f;
  for (int i = t; i < HW; i += 256) sum += __expf(rc[i] - mx);
  red[t] = sum; __syncthreads();
  for (int s = 128; s > 0; s >>= 1) { if (t < s) red[t] += red[t + s]; __syncthreads(); }
  const float inv = 1.0f / red[0];
  for (int i = t; i < HW; i += 256) {
    p[((size_t)b * NH + n) * HW + i]   = __expf(rc[i] - mx) * inv;
    sig[((size_t)b * NH + n) * HW + i] = 1.0f / (1.0f + __expf(-rd[i]));
  }
}

// ---------------------------------------------------------------------------
// K5a: vbar[b,n,c] = sum_s p[b,n,s] * v[b,c,s]
// ---------------------------------------------------------------------------
__global__ __launch_bounds__(256)
void k_vbar(const float* __restrict__ v, const float* __restrict__ p,
            float* __restrict__ vbar) {
  const int n = blockIdx.x, b = blockIdx.y, t = threadIdx.x;
  __shared__ float ps[HW];
  const float* pr = p + ((size_t)b * NH + n) * HW;
  for (int i = t; i < HW; i += 256) ps[i] = pr[i];
  __syncthreads();
  const float* vb = v + (size_t)b * CH * HW;
  for (int c = t; c < CH; c += 256) {
    const float* vr = vb + (size_t)c * HW;
    float acc = 0.f;
    for (int s = 0; s < HW; ++s) acc += ps[s] * vr[s];
    vbar[((size_t)b * NH + n) * CH + c] = acc;
  }
}

// ---------------------------------------------------------------------------
// K5b: attn[b, n*64+d] = sum_c vbar[b,n,c]*Wvs[n*64+d,c] + bvs  (softmax sums to 1)
// ---------------------------------------------------------------------------
__global__ __launch_bounds__(256)
void k_attn(const float* __restrict__ vbar, const float* __restrict__ wvs,
            const float* __restrict__ bvs, float* __restrict__ attn) {
  const int b = blockIdx.x, t = threadIdx.x;
  for (int o = t; o < CH; o += 256) {
    const int n = o >> 6;
    const float* vb = vbar + ((size_t)b * NH + n) * CH;
    const float* wr = wvs + (size_t)o * CH;
    float acc = 0.f;
    for (int c = 0; c < CH; ++c) acc += vb[c] * wr[c];
    attn[b * CH + o] = acc + bvs[o];
  }
}

// ---------------------------------------------------------------------------
// K6: fused outputs. raw[b,ch,y,x] = sig[b,n,x*32+y]*attn[b,ch]  (the permute
// transposes the spatial grid). attn_map = raw; out = leaky(BN(raw)+v, 0.1).
// ---------------------------------------------------------------------------
__global__ __launch_bounds__(256)
void k_output(const float* __restrict__ v, const float* __restrict__ sig,
              const float* __restrict__ attn,
              const float* __restrict__ gamma, const float* __restrict__ beta,
              const float* __restrict__ mean, const float* __restrict__ var,
              float* __restrict__ out, float* __restrict__ attn_map) {
  const int b = blockIdx.y;
  const size_t idx = (size_t)blockIdx.x * 256 + threadIdx.x; // 0..512*1024
  const int ch = (int)(idx >> 10);
  const int yx = (int)(idx & 1023);
  const int y = yx >> 5, x = yx & 31;
  const int n = ch >> 6;
  const float raw = sig[((size_t)b * NH + n) * HW + x * 32 + y] * attn[b * CH + ch];
  const size_t o = ((size_t)b * CH + ch) * HW + yx;
  attn_map[o] = raw;
  const float scale = gamma[ch] * rsqrtf(var[ch] + 1e-5f);
  const float val = (raw - mean[ch]) * scale + beta[ch] + v[o];
  out[o] = (val >= 0.f) ? val : 0.1f * val;
}

// ---------------------------------------------------------------------------
// K7: m_attn via factored 3x3 conv:
//   W'[n,ky,kx] = sum_d attn[b,n*64+d]*w_m[n*64+d,ky,kx]
//   m[b,y,x] = bias + sum_n sum_{ky,kx} sig[b,n,(x+kx-1)*32+(y+ky-1)]*W'
// ---------------------------------------------------------------------------
__global__ __launch_bounds__(1024)
void k_mattn(const float* __restrict__ sig, const float* __restrict__ attn,
             const float* __restrict__ wm, const float* __restrict__ wmb,
             float* __restrict__ m_attn) {
  const int b = blockIdx.x, t = threadIdx.x;
  __shared__ float Wp[NH][9];
  __shared__ float sg[NH][HW];   // s index = x*32 + y
  for (int i = t; i < NH * HW; i += 1024)
    (&sg[0][0])[i] = sig[(size_t)b * NH * HW + i];
  if (t < NH * 9) {
    const int n = t / 9, k = t % 9;
    float acc = 0.f;
    for (int d = 0; d < DK; ++d)
      acc += attn[b * CH + n * DK + d] * wm[(size_t)(n * DK + d) * 9 + k];
    Wp[n][k] = acc;
  }
  __syncthreads();
  const int y = t >> 5, x = t & 31;
  float acc = wmb[0];
#pragma unroll
  for (int n = 0; n < NH; ++n) {
    for (int ky = 0; ky < 3; ++ky) {
      const int yy = y + ky - 1;
      if (yy < 0 || yy > 31) continue;
      for (int kx = 0; kx < 3; ++kx) {
        const int xx = x + kx - 1;
        if (xx < 0 || xx > 31) continue;
        acc += sg[n][xx * 32 + yy] * Wp[n][ky * 3 + kx];
      }
    }
  }
  m_attn[b * HW + y * 32 + x] = acc;
}

// ---------------------------------------------------------------------------
extern "C" void kernel_launch(void* const* d_in, const int* in_sizes, int n_in,
                              void* d_out, int out_size, void* d_ws, size_t ws_size,
                              hipStream_t stream) {
  const float* q      = (const float*)d_in[0];
  const float* v      = (const float*)d_in[1];
  // d_in[2] = mask: unused (softmax over size-1 axis makes flap == 1 exactly)
  const float* wqs    = (const float*)d_in[3];
  const float* bqs    = (const float*)d_in[4];
  // d_in[5], d_in[6] = w_qs_att: cancels out of softmax-over-singleton
  const float* wkc    = (const float*)d_in[7];
  const float* bkc    = (const float*)d_in[8];
  const float* wkd    = (const float*)d_in[9];
  const float* bkd    = (const float*)d_in[10];
  const float* wvs    = (const float*)d_in[11];
  const float* bvs    = (const float*)d_in[12];
  const float* wm     = (const float*)d_in[13];
  const float* wmb    = (const float*)d_in[14];
  const float* gamma  = (const float*)d_in[15];
  const float* beta   = (const float*)d_in[16];
  const float* mean   = (const float*)d_in[17];
  const float* var    = (const float*)d_in[18];

  float* ws = (float*)d_ws;
  float* qh_sum = ws;                         // 32*512          = 16384
  float* U      = qh_sum + NB * CH;           // 32*16*512       = 262144
  float* kconst = U + NB * 16 * CH;           // 32*16           = 512
  float* sc     = kconst + NB * 16;           // 32*16*1024      = 524288
  float* p      = sc + (size_t)NB * 16 * HW;  // 32*8*1024       = 262144
  float* sig    = p + (size_t)NB * NH * HW;   // 32*8*1024       = 262144
  float* vbar   = sig + (size_t)NB * NH * HW; // 32*8*512        = 131072
  float* attn   = vbar + (size_t)NB * NH * CH;// 32*512          = 16384
                                              // total ~5.9 MB

  float* out      = (float*)d_out;                          // 32*512*1024
  float* m_attn   = out + (size_t)NB * CH * HW;             // 32*1024
  float* attn_map = m_attn + (size_t)NB * HW;               // 32*512*1024

  k_qsum_proj<<<dim3(NB), dim3(256), 0, stream>>>(q, wqs, bqs, qh_sum);
  k_fold<<<dim3(16, NB), dim3(128), 0, stream>>>(qh_sum, wkc, bkc, wkd, bkd, U, kconst);
  k_score_wmma<<<dim3(HW / 16, NB), dim3(32), 0, stream>>>(v, U, kconst, sc);
  k_softmax_sig<<<dim3(NH, NB), dim3(256), 0, stream>>>(sc, p, sig);
  k_vbar<<<dim3(NH, NB), dim3(256), 0, stream>>>(v, p, vbar);
  k_attn<<<dim3(NB), dim3(256), 0, stream>>>(vbar, wvs, bvs, attn);
  k_output<<<dim3((CH * HW) / 256, NB), dim3(256), 0, stream>>>(
      v, sig, attn, gamma, beta, mean, var, out, attn_map);
  k_mattn<<<dim3(NB), dim3(1024), 0, stream>>>(sig, attn, wm, wmb, m_attn);
}